// Graph_Layer_4440996184607
// MI455X (gfx1250) — compile-verified
//
#include <hip/hip_runtime.h>

typedef __attribute__((ext_vector_type(2))) float v2f;
typedef __attribute__((ext_vector_type(8))) float v8f;

#define N_ROWS  8192
#define D_SIM   512
#define IN_SIZE 1024
#define N_OUT   1024
#define EPSV    1e-12f

#define S_STRIDE 516   // 512 + 4 pad: lanes 0..15(31) hit distinct LDS banks
#define P_STRIDE 132   // 128 + 4 pad
#define JT       128   // j-tile width

// -------- Kernel 1: inverse row norms of sim_feat (with eps clamp) --------
__global__ __launch_bounds__(256)
void inv_norm_kernel(const float* __restrict__ sim, float* __restrict__ invn)
{
    const int w    = threadIdx.x >> 5;
    const int lane = threadIdx.x & 31;
    const int row  = blockIdx.x * 8 + w;
    const float* p = sim + (size_t)row * D_SIM;
    float ss = 0.f;
    #pragma unroll
    for (int i = 0; i < D_SIM / 32; ++i) {
        float v = p[lane + 32 * i];
        ss += v * v;
    }
    for (int m = 16; m >= 1; m >>= 1) ss += __shfl_xor(ss, m, 32);  // wave32
    if (lane == 0) invn[row] = 1.0f / fmaxf(sqrtf(ss), EPSV);
}

// -------- Kernel 2: fused  temp = (G @ x) / rowsum(G), G never materialized --------
// Block = 256 threads (8 wave32), owns 32 output rows and all 1024 columns.
// Every B operand (global load) feeds TWO wmma ops (M-tiles 0/1).
__global__ __launch_bounds__(256)
void fused_gx_kernel(const float* __restrict__ x, const float* __restrict__ sim,
                     const float* __restrict__ invn, float* __restrict__ temp)
{
    __shared__ float s_lds[32 * S_STRIDE];   // 32 raw sim rows (row scale cancels)
    __shared__ float p_lds[32 * P_STRIDE];   // 32x128 P tile
    __shared__ float rowsum[32];

    const int tid  = threadIdx.x;
    const int w    = tid >> 5;        // wave id 0..7
    const int lane = tid & 31;
    const int lo   = lane & 15;       // 0..15
    const int hi   = lane >> 4;       // 0..1
    const int row0 = blockIdx.x * 32;

    // cooperative stage of the 32 A-rows
    for (int i = tid; i < 32 * D_SIM; i += 256) {
        int m = i >> 9, k = i & (D_SIM - 1);
        s_lds[m * S_STRIDE + k] = sim[(size_t)(row0 + m) * D_SIM + k];
    }
    if (tid < 32) rowsum[tid] = 0.f;

    v8f acc0[8], acc1[8];             // [col tile][M-tile 0 / 1]
    #pragma unroll
    for (int t = 0; t < 8; ++t) { acc0[t] = {}; acc1[t] = {}; }

    __syncthreads();

    for (int jt = 0; jt < N_ROWS / JT; ++jt) {
        const int j0 = jt * JT;

        // ---- phase 1: wave w computes P[0:32, 16w..16w+16) = S_row . S_j^T ----
        const int   jcol = j0 + 16 * w + lo;      // this lane's N (a j-row)
        const float binv = invn[jcol];            // fold column normalization in
        const float* bsrc = sim + (size_t)jcol * D_SIM;
        v8f p0 = {}, p1 = {};
        #pragma unroll 4
        for (int k0 = 0; k0 < D_SIM; k0 += 4) {
            v2f b = *(const v2f*)&bsrc[k0 + 2 * hi];
            b[0] *= binv; b[1] *= binv;
            v2f a0 = *(const v2f*)&s_lds[lo * S_STRIDE + k0 + 2 * hi];
            v2f a1 = *(const v2f*)&s_lds[(16 + lo) * S_STRIDE + k0 + 2 * hi];
            p0 = __builtin_amdgcn_wmma_f32_16x16x4_f32(false, a0, false, b,
                                                       (short)0, p0, false, false);
            p1 = __builtin_amdgcn_wmma_f32_16x16x4_f32(false, a1, false, b,
                                                       (short)0, p1, false, false);
        }
        #pragma unroll
        for (int r = 0; r < 8; ++r) {
            p_lds[(r + 8 * hi) * P_STRIDE + 16 * w + lo]        = p0[r];
            p_lds[(16 + r + 8 * hi) * P_STRIDE + 16 * w + lo]   = p1[r];
        }
        __syncthreads();

        // running rowsum (wave 0; rows contiguous in LDS -> wide ds loads)
        if (w == 0) {
            float rs = 0.f;
            #pragma unroll 8
            for (int n = 0; n < JT; ++n) rs += p_lds[lane * P_STRIDE + n];
            rowsum[lane] += rs;
        }

        // ---- phase 2: acc[32 x 128w..] += P(32x128) @ x(128x128) ----
        #pragma unroll 2
        for (int kj = 0; kj < JT; kj += 4) {
            v2f a0 = *(const v2f*)&p_lds[lo * P_STRIDE + kj + 2 * hi];
            v2f a1 = *(const v2f*)&p_lds[(16 + lo) * P_STRIDE + kj + 2 * hi];
            const float* xrow = x + (size_t)(j0 + kj + 2 * hi) * IN_SIZE;
            #pragma unroll
            for (int ct = 0; ct < 8; ++ct) {
                const int col = 128 * w + 16 * ct + lo;
                v2f b; b[0] = xrow[col]; b[1] = xrow[IN_SIZE + col];
                acc0[ct] = __builtin_amdgcn_wmma_f32_16x16x4_f32(false, a0, false, b,
                                                     (short)0, acc0[ct], false, false);
                acc1[ct] = __builtin_amdgcn_wmma_f32_16x16x4_f32(false, a1, false, b,
                                                     (short)0, acc1[ct], false, false);
            }
        }
        __syncthreads();   // protect P (and rowsum order) before next tile
    }

    // ---- finalize: divide by rowsum, store temp ----
    float rinv0[8], rinv1[8];
    #pragma unroll
    for (int r = 0; r < 8; ++r) {
        rinv0[r] = 1.0f / rowsum[r + 8 * hi];
        rinv1[r] = 1.0f / rowsum[16 + r + 8 * hi];
    }
    for (int ct = 0; ct < 8; ++ct) {
        const int col = 128 * w + 16 * ct + lo;
        #pragma unroll
        for (int r = 0; r < 8; ++r) {
            temp[(size_t)(row0 + r + 8 * hi) * IN_SIZE + col]      = acc0[ct][r] * rinv0[r];
            temp[(size_t)(row0 + 16 + r + 8 * hi) * IN_SIZE + col] = acc1[ct][r] * rinv1[r];
        }
    }
}

// -------- Kernel 3: out = temp @ weight  (fp32 WMMA GEMM) --------
// Block = 128 threads (4 waves); each wave computes a 32x64 tile,
// sharing each B load across two M-tiles.
__global__ __launch_bounds__(128)
void gemm_out_kernel(const float* __restrict__ temp, const float* __restrict__ wgt,
                     float* __restrict__ out)
{
    const int tid  = threadIdx.x;
    const int w    = tid >> 5;
    const int lane = tid & 31;
    const int lo   = lane & 15;
    const int hi   = lane >> 4;
    const int row0 = blockIdx.x * 32;
    const int col0 = blockIdx.y * 256 + w * 64;

    v8f acc0[4], acc1[4];
    #pragma unroll
    for (int t = 0; t < 4; ++t) { acc0[t] = {}; acc1[t] = {}; }

    for (int k0 = 0; k0 < IN_SIZE; k0 += 4) {
        v2f a0 = *(const v2f*)&temp[(size_t)(row0 + lo) * IN_SIZE + k0 + 2 * hi];
        v2f a1 = *(const v2f*)&temp[(size_t)(row0 + 16 + lo) * IN_SIZE + k0 + 2 * hi];
        const float* wrow = wgt + (size_t)(k0 + 2 * hi) * N_OUT;
        #pragma unroll
        for (int t = 0; t < 4; ++t) {
            const int col = col0 + 16 * t + lo;
            v2f b; b[0] = wrow[col]; b[1] = wrow[N_OUT + col];
            acc0[t] = __builtin_amdgcn_wmma_f32_16x16x4_f32(false, a0, false, b,
                                                 (short)0, acc0[t], false, false);
            acc1[t] = __builtin_amdgcn_wmma_f32_16x16x4_f32(false, a1, false, b,
                                                 (short)0, acc1[t], false, false);
        }
    }
    #pragma unroll
    for (int t = 0; t < 4; ++t) {
        const int col = col0 + 16 * t + lo;
        #pragma unroll
        for (int r = 0; r < 8; ++r) {
            out[(size_t)(row0 + r + 8 * hi) * N_OUT + col]      = acc0[t][r];
            out[(size_t)(row0 + 16 + r + 8 * hi) * N_OUT + col] = acc1[t][r];
        }
    }
}

extern "C" void kernel_launch(void* const* d_in, const int* in_sizes, int n_in,
                              void* d_out, int out_size, void* d_ws, size_t ws_size,
                              hipStream_t stream) {
    const float* x   = (const float*)d_in[0];   // [8192, 1024]
    const float* sim = (const float*)d_in[1];   // [8192, 512]
    const float* wgt = (const float*)d_in[2];   // [1024, 1024]
    float* out  = (float*)d_out;                // [8192, 1024]
    float* temp = (float*)d_ws;                                   // 32 MB
    float* invn = temp + (size_t)N_ROWS * IN_SIZE;                // +32 KB

    inv_norm_kernel<<<N_ROWS / 8, 256, 0, stream>>>(sim, invn);
    fused_gx_kernel<<<N_ROWS / 32, 256, 0, stream>>>(x, sim, invn, temp);
    dim3 gc(N_ROWS / 32, N_OUT / 256);
    gemm_out_kernel<<<gc, 128, 0, stream>>>(temp, wgt, out);
}